// TemperatureScaler_26955214750130
// MI455X (gfx1250) — compile-verified
//
#include <hip/hip_runtime.h>

// Piecewise-linear temperature scaler (mode='absolute', pieces=8, n_temp=1).
//   bin  = #{ j : thr_sorted[j] < x }            (searchsorted side='left')
//   out  = T[bin] * (x - thr_pad[bin]) + c_pad[bin]
//
// Memory-bound: 65.5 MB traffic @ 23.3 TB/s -> ~2.8us floor.
//  - 4x float4 per thread: four b128 loads issued back-to-back per wave
//    (clause-able) for deep memory-level parallelism.
//  - Non-temporal TH on loads/stores: single-pass stream, don't pollute L2.
//  - Per-block LDS table (T, thr_pad, c_pad) gathered with conflict-free
//    ds_load_b128 (8 distinct 16B entries -> distinct bank groups).

#define PIECES 8
#define NTHR   7
#define TEMP_CLAMP_MIN 1e-4f
#define VEC_PER_THREAD 4

typedef float vfloat4 __attribute__((ext_vector_type(4)));

__device__ __forceinline__ float apply_piece(float x, const float4* __restrict__ tbl,
                                             float t0, float t1, float t2, float t3,
                                             float t4, float t5, float t6) {
    // searchsorted side='left': count of thresholds strictly below x
    int b = (x > t0) + (x > t1) + (x > t2) + (x > t3) + (x > t4) + (x > t5) + (x > t6);
    float4 p = tbl[b];                       // ds_load_b128 gather
    return fmaf(p.x, x - p.y, p.z);          // T*(x - thr_pad) + c_pad
}

__global__ __launch_bounds__(256) void TemperatureScaler_kernel(
    const float* __restrict__ logits,
    const float* __restrict__ temperature,   // (1, 8)
    const float* __restrict__ thresholds,    // (1, 7)
    float* __restrict__ out,
    int n)
{
    __shared__ float4 tbl[PIECES];   // (T[b], thr_pad[b], c_pad[b], 0)
    __shared__ float  sthr[NTHR];    // sorted thresholds for bin search

    if (threadIdx.x == 0) {
        float T[PIECES];
        #pragma unroll
        for (int i = 0; i < PIECES; ++i)
            T[i] = fmaxf(temperature[i], TEMP_CLAMP_MIN);  // jnp.clip(temp, 1e-4)

        float thr[NTHR];
        #pragma unroll
        for (int i = 0; i < NTHR; ++i) thr[i] = thresholds[i];

        // jnp.sort(thresholds) — 7-element insertion sort (thread 0 only)
        for (int i = 1; i < NTHR; ++i) {
            float v = thr[i];
            int j = i - 1;
            while (j >= 0 && thr[j] > v) { thr[j + 1] = thr[j]; --j; }
            thr[j + 1] = v;
        }

        // c = cumsum(diff(pad(thr, left 0)) * T[:-1])
        float c[NTHR];
        float prev = 0.0f, acc = 0.0f;
        #pragma unroll
        for (int j = 0; j < NTHR; ++j) {
            acc += (thr[j] - prev) * T[j];
            c[j] = acc;
            prev = thr[j];
        }

        tbl[0] = make_float4(T[0], 0.0f, 0.0f, 0.0f);      // bin 0: T0 * x
        #pragma unroll
        for (int b = 1; b < PIECES; ++b)
            tbl[b] = make_float4(T[b], thr[b - 1], c[b - 1], 0.0f);
        #pragma unroll
        for (int j = 0; j < NTHR; ++j) sthr[j] = thr[j];
    }
    __syncthreads();

    // Broadcast threshold reads (LDS broadcast, then live in VGPRs)
    const float t0 = sthr[0], t1 = sthr[1], t2 = sthr[2], t3 = sthr[3];
    const float t4 = sthr[4], t5 = sthr[5], t6 = sthr[6];

    const int nvec     = n >> 2;                              // float4 count
    const int tid      = (int)(blockIdx.x * blockDim.x + threadIdx.x);
    const int nthreads = (int)(gridDim.x * blockDim.x);
    const vfloat4* __restrict__ in4  = reinterpret_cast<const vfloat4*>(logits);
    vfloat4* __restrict__       out4 = reinterpret_cast<vfloat4*>(out);

    // Each super-iteration: 4 coalesced b128 NT loads in flight, then 4 NT stores.
    for (int base = tid; base < nvec; base += nthreads * VEC_PER_THREAD) {
        vfloat4 x[VEC_PER_THREAD];
        #pragma unroll
        for (int k = 0; k < VEC_PER_THREAD; ++k) {
            int i = base + k * nthreads;
            if (i < nvec) x[k] = __builtin_nontemporal_load(in4 + i);
        }
        #pragma unroll
        for (int k = 0; k < VEC_PER_THREAD; ++k) {
            int i = base + k * nthreads;
            if (i < nvec) {
                vfloat4 r;
                r.x = apply_piece(x[k].x, tbl, t0, t1, t2, t3, t4, t5, t6);
                r.y = apply_piece(x[k].y, tbl, t0, t1, t2, t3, t4, t5, t6);
                r.z = apply_piece(x[k].z, tbl, t0, t1, t2, t3, t4, t5, t6);
                r.w = apply_piece(x[k].w, tbl, t0, t1, t2, t3, t4, t5, t6);
                __builtin_nontemporal_store(r, out4 + i);
            }
        }
    }

    // Scalar tail (n % 4 != 0); for B*V = 8,192,000 this never executes.
    const int tail = nvec << 2;
    for (int i = tail + tid; i < n; i += nthreads) {
        out[i] = apply_piece(logits[i], tbl, t0, t1, t2, t3, t4, t5, t6);
    }
}

extern "C" void kernel_launch(void* const* d_in, const int* in_sizes, int n_in,
                              void* d_out, int out_size, void* d_ws, size_t ws_size,
                              hipStream_t stream) {
    (void)n_in; (void)d_ws; (void)ws_size; (void)out_size;
    const float* logits      = (const float*)d_in[0];
    const float* temperature = (const float*)d_in[1];
    const float* thresholds  = (const float*)d_in[2];
    float*       out         = (float*)d_out;

    const int n       = in_sizes[0];              // 64 * 128000 = 8,192,000
    const int nvec    = n >> 2;                   // 2,048,000 float4s
    const int threads = 256;                      // 8 wave32 waves / block
    const int per_blk = threads * VEC_PER_THREAD; // 1024 float4s / block
    int blocks = (nvec + per_blk - 1) / per_blk;  // 2000 blocks, exact tiling
    if (blocks < 1) blocks = 1;
    if (blocks > 65535) blocks = 65535;

    TemperatureScaler_kernel<<<blocks, threads, 0, stream>>>(
        logits, temperature, thresholds, out, n);
}